// DownSampleBlock_17463337026271
// MI455X (gfx1250) — compile-verified
//
#include <hip/hip_runtime.h>
#include <hip/hip_bf16.h>

typedef float v2f __attribute__((ext_vector_type(2)));
typedef float v8f __attribute__((ext_vector_type(8)));

// Problem sizes (fixed by the reference's setup_inputs)
#define BB    8
#define NN    2048
#define CC    256
#define MM    (BB * NN)     // 16384
#define KOUT  1024

#define LDS_STRIDE 34       // even (8B-aligned b64 loads) and co-prime-ish with 64 banks

// ---------------------------------------------------------------------------
// 1) LayerNorm with transposed read: features (B,C,N) -> f (B*N, C)
//    one wave (32 lanes) per point, 8 channels per lane
// ---------------------------------------------------------------------------
__global__ __launch_bounds__(256) void ln_kernel(const float* __restrict__ feat,
                                                 const float* __restrict__ gamma,
                                                 const float* __restrict__ beta,
                                                 float* __restrict__ f) {
  int wave = threadIdx.x >> 5, lane = threadIdx.x & 31;
  int row = blockIdx.x * 8 + wave;          // [0, B*N)
  int b = row >> 11, n = row & (NN - 1);
  const float* src = feat + (size_t)b * CC * NN + n;
  float x[8];
  float s = 0.f;
#pragma unroll
  for (int j = 0; j < 8; ++j) {
    x[j] = src[(size_t)(lane + 32 * j) * NN];
    s += x[j];
  }
#pragma unroll
  for (int off = 16; off; off >>= 1) s += __shfl_xor(s, off, 32);
  float mu = s * (1.0f / 256.0f);
  float vs = 0.f;
#pragma unroll
  for (int j = 0; j < 8; ++j) { float d = x[j] - mu; vs += d * d; }
#pragma unroll
  for (int off = 16; off; off >>= 1) vs += __shfl_xor(vs, off, 32);
  float rs = 1.0f / sqrtf(vs * (1.0f / 256.0f) + 1e-6f);
  float* dst = f + (size_t)row * CC;
#pragma unroll
  for (int j = 0; j < 8; ++j) {
    int c = lane + 32 * j;
    dst[c] = (x[j] - mu) * rs * gamma[c] + beta[c];
  }
}

// ---------------------------------------------------------------------------
// 2) f32 WMMA GEMM:  Out(M,256) = act(A(M,256) @ W(256,256) + bias)
//    V_WMMA_F32_16X16X4_F32, wave -> 16x64 output strip (4 accumulators).
//    W staged through LDS in 32-row K slices, stored K-transposed so each
//    lane's B pair {W[k,col], W[k+1,col]} is one aligned ds_load_b64.
//    A layout (16x4 f32): lanes 0-15 = rows, K={k,k+1}; lanes 16-31 K={k+2,k+3}
//    B layout (4x16 f32): lanes 0-15 = cols, K={k,k+1}; lanes 16-31 K={k+2,k+3}
//    D layout: VGPR r -> M = r + 8*(lane>=16), N = lane&15
// ---------------------------------------------------------------------------
__global__ __launch_bounds__(256) void gemm_wmma(const float* __restrict__ A,
                                                 const float* __restrict__ W,
                                                 const float* __restrict__ bias,
                                                 float* __restrict__ Out,
                                                 int relu) {
  __shared__ float WldsT[256 * LDS_STRIDE];  // ~34 KB, [col][k] transposed slice
  int tid  = threadIdx.x;
  int wave = tid >> 5, lane = tid & 31;
  int half = lane & 15, hi = lane >> 4;
  int mTile = blockIdx.x * 2 + (wave >> 2); // 2 M-tiles per block
  int nBase = (wave & 3) * 64;              // 4 waves span 256 cols
  int row = mTile * 16 + half;
  const float* Arow = A + (size_t)row * 256;

  v8f acc[4];
#pragma unroll
  for (int t = 0; t < 4; ++t) acc[t] = (v8f){0.f,0.f,0.f,0.f,0.f,0.f,0.f,0.f};

  for (int ks = 0; ks < 256; ks += 32) {
    // Fill: thread tid owns column `tid`; coalesced global reads,
    // contiguous per-thread LDS writes (bank-conflict-free: gcd(34,64)=2,
    // lane*34 mod 64 has period 32).
#pragma unroll 8
    for (int i = 0; i < 32; ++i)
      WldsT[tid * LDS_STRIDE + i] = W[(size_t)(ks + i) * 256 + tid];
    __syncthreads();

#pragma unroll
    for (int kk = 0; kk < 32; kk += 4) {
      int kl = kk + hi * 2;                 // this lane's K pair within slice
      v2f a = *(const v2f*)(Arow + ks + kl);
#pragma unroll
      for (int t = 0; t < 4; ++t) {
        int col = nBase + t * 16 + half;
        v2f bf = *(const v2f*)(&WldsT[col * LDS_STRIDE + kl]);  // one ds_load_b64
        acc[t] = __builtin_amdgcn_wmma_f32_16x16x4_f32(
            false, a, false, bf, (short)0, acc[t], false, false);
      }
    }
    __syncthreads();
  }

#pragma unroll
  for (int t = 0; t < 4; ++t) {
    int col = nBase + t * 16 + half;
    float bv = bias[col];
#pragma unroll
    for (int r = 0; r < 8; ++r) {
      int m = mTile * 16 + r + hi * 8;
      float v = acc[t][r] + bv;
      if (relu) v = fmaxf(v, 0.0f);
      Out[(size_t)m * 256 + col] = v;
    }
  }
}

// ---------------------------------------------------------------------------
// 3) Score GEMV: w[row] = dot(H[row,:], Wd2) + bd2 ; one wave per row
// ---------------------------------------------------------------------------
__global__ __launch_bounds__(256) void score_kernel(const float* __restrict__ H,
                                                    const float* __restrict__ wvec,
                                                    const float* __restrict__ bsc,
                                                    float* __restrict__ w) {
  int wave = threadIdx.x >> 5, lane = threadIdx.x & 31;
  int row = blockIdx.x * 8 + wave;
  const float* h = H + (size_t)row * 256;
  float s = 0.f;
#pragma unroll
  for (int j = 0; j < 8; ++j) {
    int c = j * 32 + lane;
    s += h[c] * wvec[c];
  }
#pragma unroll
  for (int off = 16; off; off >>= 1) s += __shfl_xor(s, off, 32);
  if (lane == 0) w[row] = s + bsc[0];
}

// ---------------------------------------------------------------------------
// 4) Per-batch bitonic sort of 2048 (value desc, index asc on ties).
//    First 1024 sorted entries == continuous_topk selection order.
// ---------------------------------------------------------------------------
__global__ __launch_bounds__(1024) void topk_sort(const float* __restrict__ w,
                                                  int* __restrict__ sortedIdx,
                                                  int* __restrict__ idxOut) {
  __shared__ float sv[2048];
  __shared__ int   si[2048];
  int b = blockIdx.x, tid = threadIdx.x;
  for (int i = tid; i < 2048; i += 1024) { sv[i] = w[b * 2048 + i]; si[i] = i; }
  __syncthreads();
  for (int k = 2; k <= 2048; k <<= 1) {
    for (int j = k >> 1; j > 0; j >>= 1) {
      for (int i = tid; i < 2048; i += 1024) {
        int p = i ^ j;
        if (p > i) {
          bool up = ((i & k) == 0);
          float va = sv[i], vb = sv[p];
          int ia = si[i], ib = si[p];
          // "a belongs after b" in (desc value, asc index) order
          bool aAfter = (va < vb) || (va == vb && ia > ib);
          if (up == aAfter) { sv[i] = vb; sv[p] = va; si[i] = ib; si[p] = ia; }
        }
      }
      __syncthreads();
    }
  }
  for (int i = tid; i < KOUT; i += 1024) {
    sortedIdx[b * KOUT + i] = si[i];
    idxOut[b * KOUT + i] = si[i];
  }
}

// ---------------------------------------------------------------------------
// 5) Gathers (khot values are numerically 1 at the selected index, 0 elsewhere)
// ---------------------------------------------------------------------------
__global__ __launch_bounds__(256) void gather_xyz(const float* __restrict__ xyzs,
                                                  const int* __restrict__ sortedIdx,
                                                  float* __restrict__ xout) {
  int t = blockIdx.x * 256 + threadIdx.x;   // over B*K = 8192
  if (t >= BB * KOUT) return;
  int b = t >> 10;
  int idx = sortedIdx[t];
  const float* s = xyzs + ((size_t)b * NN + idx) * 3;
  float* d = xout + (size_t)t * 3;
  d[0] = s[0]; d[1] = s[1]; d[2] = s[2];
}

__global__ __launch_bounds__(256) void gather_feats(const float* __restrict__ new_f,
                                                    const int* __restrict__ sortedIdx,
                                                    float* __restrict__ fout) {
  int b = blockIdx.x >> 8, c = blockIdx.x & 255;   // grid = B*C = 2048
  for (int k = threadIdx.x; k < KOUT; k += 256) {
    int idx = sortedIdx[b * KOUT + k];
    fout[((size_t)b * CC + c) * KOUT + k] =
        new_f[((size_t)b * NN + idx) * CC + c];
  }
}

// ---------------------------------------------------------------------------
extern "C" void kernel_launch(void* const* d_in, const int* in_sizes, int n_in,
                              void* d_out, int out_size, void* d_ws, size_t ws_size,
                              hipStream_t stream) {
  const float* xyzs  = (const float*)d_in[0];
  const float* feats = (const float*)d_in[1];
  const float* gamma = (const float*)d_in[2];
  const float* beta  = (const float*)d_in[3];
  const float* Wu1   = (const float*)d_in[4];
  const float* bu1   = (const float*)d_in[5];
  const float* Wu2   = (const float*)d_in[6];
  const float* bu2   = (const float*)d_in[7];
  const float* Wd1   = (const float*)d_in[8];
  const float* bd1   = (const float*)d_in[9];
  const float* Wd2   = (const float*)d_in[10];
  const float* bd2   = (const float*)d_in[11];

  // Workspace layout (floats)
  float* f    = (float*)d_ws;                 // 16 MB
  float* tmp  = f   + (size_t)MM * CC;        // 16 MB (reused by both branches)
  float* nf   = tmp + (size_t)MM * CC;        // 16 MB
  float* wsc  = nf  + (size_t)MM * CC;        // 64 KB scores
  int*   sidx = (int*)(wsc + MM);             // B*K ints

  // Output layout: xyzs_out (B,K,3) | feats (B,256,K) | indices (B,K) int32
  float* xout   = (float*)d_out;
  float* foutp  = xout + (size_t)BB * KOUT * 3;
  int*   idxOut = (int*)(foutp + (size_t)BB * CC * KOUT);

  ln_kernel<<<MM / 8, 256, 0, stream>>>(feats, gamma, beta, f);

  gemm_wmma<<<MM / 32, 256, 0, stream>>>(f, Wu1, bu1, tmp, 1);    // relu
  gemm_wmma<<<MM / 32, 256, 0, stream>>>(tmp, Wu2, bu2, nf, 0);   // new_f
  gemm_wmma<<<MM / 32, 256, 0, stream>>>(f, Wd1, bd1, tmp, 1);    // relu branch
  score_kernel<<<MM / 8, 256, 0, stream>>>(tmp, Wd2, bd2, wsc);   // w (B,N)

  topk_sort<<<BB, 1024, 0, stream>>>(wsc, sidx, idxOut);

  gather_xyz<<<(BB * KOUT + 255) / 256, 256, 0, stream>>>(xyzs, sidx, xout);
  gather_feats<<<BB * CC, 256, 0, stream>>>(nf, sidx, foutp);
}